// LogReg_8151847928094
// MI455X (gfx1250) — compile-verified
//
#include <hip/hip_runtime.h>

typedef __attribute__((ext_vector_type(2))) float v2f;
typedef __attribute__((ext_vector_type(8))) float v8f;

#define SEQ 200
#define BATCH 2048
#define NCHUNK (SEQ / 4)       // 50 chunks of 4 sequence rows
#define NWAVE 5                // 5 waves x 10 chunks = 50, no tail
#define CHUNKS_PER_WAVE (NCHUNK / NWAVE)

// One block = one tile of 16 batch columns. 5 waves split the 50 K-chunks
// into fixed-size contiguous ranges of 10 (fully unrollable, no divergence,
// no tail -> EXEC is provably all-ones at every WMMA).
// Each wave accumulates D += ones(16x4) * B(4x16) with full-fp32 WMMA, where
// B[k][n] = w[text[s+k, bbase+n]]. Every row of D equals the column sums, so
// the result does not depend on the exact K<->(vgpr,half) mapping of B --
// only N = lane%16 (fixed by the documented C/D layout) matters.
__global__ __launch_bounds__(NWAVE * 32) void bow_linear_wmma(
    const int* __restrict__ text,   // [SEQ, BATCH] token ids
    const float* __restrict__ w,    // [V] weights
    const float* __restrict__ bias, // [1]
    float* __restrict__ out) {      // [BATCH]
  __shared__ float partial[NWAVE][16];

  const int lane = threadIdx.x & 31;
  const int wave = threadIdx.x >> 5;
  const int n    = lane & 15;   // output column within tile (N = lane%16)
  const int half = lane >> 4;   // lane half selects which K pair this lane holds
  const int col  = blockIdx.x * 16 + n;

  v2f a;
  a.x = 1.0f;                   // A = all-ones: D[m][n] = sum_k B[k][n]
  a.y = 1.0f;
  v8f c = {};                   // fp32 accumulator (C/D, 8 VGPRs)

  const int base = wave * CHUNKS_PER_WAVE;  // contiguous chunk range per wave

#pragma unroll
  for (int i = 0; i < CHUNKS_PER_WAVE; ++i) {
    const int s  = (base + i) * 4;
    const int r0 = s + 2 * half;      // K = 0 + 2*half  (B vgpr 0)
    const int r1 = s + 1 + 2 * half;  // K = 1 + 2*half  (B vgpr 1)
    const int t0 = text[(size_t)r0 * BATCH + col];  // coalesced across lanes
    const int t1 = text[(size_t)r1 * BATCH + col];
    v2f bm;
    bm.x = w[t0];                     // gather from L2-resident 400KB table
    bm.y = w[t1];
    // D = A*B + C, full fp32: v_wmma_f32_16x16x4_f32
    c = __builtin_amdgcn_wmma_f32_16x16x4_f32(
        /*neg_a=*/false, a, /*neg_b=*/false, bm,
        /*c_mod=*/(short)0, c, /*reuse_a=*/false, /*reuse_b=*/false);
  }

  // D vgpr0, lanes 0-15: M=0, N=lane -> per-column partial sum for this wave.
  if (half == 0) partial[wave][n] = c[0];
  __syncthreads();

  if (wave == 0 && half == 0) {
    float r = partial[0][n] + partial[1][n] + partial[2][n] + partial[3][n] +
              partial[4][n] + bias[0];
    out[col] = r;   // deterministic combine: no atomics, replay-safe
  }
}

extern "C" void kernel_launch(void* const* d_in, const int* in_sizes, int n_in,
                              void* d_out, int out_size, void* d_ws,
                              size_t ws_size, hipStream_t stream) {
  const int*   text = (const int*)d_in[0];   // [SEQ*BATCH] token ids
  const float* w    = (const float*)d_in[1]; // [V]
  const float* bias = (const float*)d_in[2]; // [1]
  float*       out  = (float*)d_out;         // [BATCH]

  (void)in_sizes; (void)n_in; (void)out_size; (void)d_ws; (void)ws_size;

  dim3 grid(BATCH / 16);     // 128 tiles of 16 batch columns
  dim3 block(NWAVE * 32);    // 5 waves per tile, 10 chunks each
  bow_linear_wmma<<<grid, block, 0, stream>>>(text, w, bias, out);
}